// RPN_37117107372612
// MI455X (gfx1250) — compile-verified
//
#include <hip/hip_runtime.h>
#include <stdint.h>

// ---------------- problem constants (match reference) ----------------
#define NUM_PRE    2000
#define NUM_POST   1000
#define NMS_THR    0.7f
#define MIN_SIZE   0.01f
#define BBOX_CLAMP 4.135166556742356f   // log(1000/16)
#define NBATCH     16
#define NLEV       4
#define NPROP      8000                 // 4 levels * 2000
#define NEGV       (-1e30f)

// monotonic float -> uint key (larger float => larger key)
__device__ __forceinline__ unsigned f2key(float f) {
  unsigned u = __float_as_uint(f);
  return (u & 0x80000000u) ? ~u : (u | 0x80000000u);
}

// ---------------- CDNA5 async global->LDS (ASYNCcnt path) ----------------
// GVS mode: mem = SGPR64 base + VGPR 32-bit byte offset; VDST = LDS byte addr.
__device__ __forceinline__ void async_g2l_b32(uint32_t lds_byte_off,
                                              const void* sbase,
                                              uint32_t gbyte_off) {
  asm volatile("global_load_async_to_lds_b32 %0, %1, %2"
               :: "v"(lds_byte_off), "v"(gbyte_off), "s"(sbase)
               : "memory");
}
__device__ __forceinline__ void wait_async0() {
  asm volatile("s_wait_asynccnt 0" ::: "memory");
}

// ---------------- wave32 reductions ----------------
__device__ __forceinline__ void wave_argmax(float& v, int& i) {
#pragma unroll
  for (int off = 16; off > 0; off >>= 1) {
    float ov = __shfl_down(v, off, 32);
    int   oi = __shfl_down(i, off, 32);
    if (ov > v || (ov == v && oi < i)) { v = ov; i = oi; }
  }
}
__device__ __forceinline__ float wave_max(float v) {
#pragma unroll
  for (int off = 16; off > 0; off >>= 1)
    v = fmaxf(v, __shfl_down(v, off, 32));
  return v;
}

// per-level pointer bundle passed by value
struct LevelArgs {
  const float* cls[NLEV];
  const float* regs[NLEV];
  const float* anch[NLEV];
  int hwshift[NLEV];   // log2(H*H)
};

// ============================================================================
// Stage A: per (image, level) exact top-2000 via 4-pass radix select, then
// compact + decode boxes for the winners only.
// grid = 64 blocks: b = blockIdx.x >> 2, level = blockIdx.x & 3.
// All cls/regs traffic is channel-major => fully coalesced linear reads.
// ============================================================================
__global__ __launch_bounds__(1024)
void rpn_topk_decode(LevelArgs la,
                     float* __restrict__ sc_ws,
                     float* __restrict__ box_ws) {
  const int lev = blockIdx.x & 3;
  const int b   = blockIdx.x >> 2;
  const int tid = threadIdx.x;
  const int bs  = blockDim.x;

  const int hwshift = la.hwshift[lev];
  const int HW = 1 << hwshift;
  const int N  = 3 * HW;
  const float* clsb = la.cls[lev] + (size_t)b * 3 * HW;

  __shared__ unsigned hist[256];
  __shared__ unsigned sh_sel, sh_K, cntG, cntE;

  // ---- radix select: find threshold key T of the 2000th largest score ----
  unsigned prefix = 0, pmask = 0;
  int K = NUM_PRE;
  for (int p = 0; p < 4; ++p) {
    const int shift = 24 - 8 * p;
    for (int i = tid; i < 256; i += bs) hist[i] = 0;
    __syncthreads();
    // ordering-free: stream scores linearly (coalesced)
    for (int i = tid; i < N; i += bs) {
      __builtin_prefetch(&clsb[i + 4 * bs], 0, 0);   // global_prefetch_b8
      unsigned key = f2key(clsb[i]);
      if ((key & pmask) == prefix)
        atomicAdd(&hist[(key >> shift) & 255u], 1u);
    }
    __syncthreads();
    if (tid == 0) {
      unsigned cum = 0; int sel = 0;
      for (int bkt = 255; bkt >= 0; --bkt) {
        unsigned c = hist[bkt];
        if (cum + c >= (unsigned)K) { sel = bkt; break; }
        cum += c;
      }
      sh_sel = (unsigned)sel;
      sh_K   = (unsigned)K - cum;
    }
    __syncthreads();
    prefix |= (sh_sel << shift);
    pmask  |= (0xFFu << shift);
    K = (int)sh_K;
    __syncthreads();
  }
  const unsigned T   = prefix;              // exact 32-bit threshold key
  const int     Krem = K;                   // how many == T to take
  const int     G    = NUM_PRE - Krem;      // count of keys strictly > T

  if (tid == 0) { cntG = 0; cntE = 0; }
  __syncthreads();

  // ---- compact winners, gather regs/anchors, decode boxes ----
  const float* rb   = la.regs[lev] + (size_t)b * 12 * HW;
  const float* anch = la.anch[lev];
  for (int i = tid; i < N; i += bs) {       // linear (coalesced) re-read
    const float s = clsb[i];
    const unsigned key = f2key(s);
    int slot = -1;
    if (key > T) {
      slot = (int)atomicAdd(&cntG, 1u);                // [0, G)
    } else if (key == T) {
      unsigned e = atomicAdd(&cntE, 1u);
      if (e < (unsigned)Krem) slot = G + (int)e;       // [G, 2000)
    }
    if (slot >= 0) {
      const int a  = i >> hwshift;          // anchor-aspect channel
      const int hw = i & (HW - 1);          // spatial index h*W+w
      const int j  = hw * 3 + a;            // (h,w,a)-flattened proposal id
      // regs permute(0,2,3,1): channel = 4*a + c at spatial hw (coalesced)
      float dx = rb[(4 * a + 0) * HW + hw];
      float dy = rb[(4 * a + 1) * HW + hw];
      float dh = rb[(4 * a + 2) * HW + hw];
      float dw = rb[(4 * a + 3) * HW + hw];
      float ax1 = anch[4 * j + 0], ay1 = anch[4 * j + 1];
      float ax2 = anch[4 * j + 2], ay2 = anch[4 * j + 3];
      float aw = ax2 - ax1, ah = ay2 - ay1;
      float cx = aw * 0.5f,  cy = ah * 0.5f;   // reference quirk: ignores anchor ctr
      dw = fminf(dw, BBOX_CLAMP);
      dh = fminf(dh, BBOX_CLAMP);
      float px = cx + dx * aw, py = cy + dy * ah;
      float pw = expf(dw) * aw, ph = expf(dh) * ah;

      const int o = b * NPROP + lev * NUM_PRE + slot;
      sc_ws[o] = s;
      float* bo = box_ws + (size_t)o * 4;
      bo[0] = px - pw * 0.5f;
      bo[1] = py - ph * 0.5f;
      bo[2] = px + pw * 0.5f;
      bo[3] = py + ph * 0.5f;
    }
  }
}

// ============================================================================
// Stage B: per-image greedy NMS, 1000 sequential steps.
// One block per image; the full 8000-proposal set (scores + SoA boxes,
// 160 KB) lives in CDNA5's big per-WGP LDS, staged via async global->LDS.
// Dynamic-LDS float layout:
//   [0,8000)   S      scores
//   [8000..)   X1 Y1 X2 Y2   (8000 each)
//   [40000,40032)  redV   [40032,40064) redI   [40064..) ctrl{bestV,bestI,pmax}
// ============================================================================
__global__ __launch_bounds__(1024)
void rpn_nms(const float* __restrict__ sc_ws,
             const float* __restrict__ box_ws,
             const int* __restrict__ img_h,
             const int* __restrict__ img_w,
             float* __restrict__ out) {
  extern __shared__ float sm[];
  float* S    = sm;
  float* X1   = sm + NPROP;
  float* Y1   = sm + 2 * NPROP;
  float* X2   = sm + 3 * NPROP;
  float* Y2   = sm + 4 * NPROP;
  float* redV = sm + 5 * NPROP;                 // 32 floats
  int*   redI = (int*)(sm + 5 * NPROP + 32);    // 32 ints
  float* ctrl = sm + 5 * NPROP + 64;            // bestV, bestI(bits), pmax

  const int b    = blockIdx.x;
  const int tid  = threadIdx.x;
  const int bs   = blockDim.x;
  const int wid  = tid >> 5, lane = tid & 31;
  const int nw   = bs >> 5;

  const float* scg = sc_ws + (size_t)b * NPROP;
  const float* bxg = box_ws + (size_t)b * NPROP * 4;

  // ---- async-stage scores + boxes (AoS -> SoA during the copy) ----
  for (int j = tid; j < NPROP; j += bs) {
    async_g2l_b32((uint32_t)(j * 4), scg, (uint32_t)(j * 4));
#pragma unroll
    for (int c = 0; c < 4; ++c)
      async_g2l_b32((uint32_t)(((c + 1) * NPROP + j) * 4), bxg,
                    (uint32_t)((j * 4 + c) * 4));
  }
  wait_async0();
  __syncthreads();

  // ---- pmax = max over all raw box coords (reference: p.max()) ----
  float m = NEGV;
  for (int j = tid; j < NPROP; j += bs)
    m = fmaxf(m, fmaxf(fmaxf(X1[j], Y1[j]), fmaxf(X2[j], Y2[j])));
  m = wave_max(m);
  if (lane == 0) redV[wid] = m;
  __syncthreads();
  if (wid == 0) {
    float v = (lane < nw) ? redV[lane] : NEGV;
    v = wave_max(v);
    if (lane == 0) ctrl[2] = v;
  }
  __syncthreads();
  const float pmax = ctrl[2];
  const float Wimg = (float)img_w[0];
  const float Himg = (float)img_h[0];

  // ---- min-size keep test (on clipped coords) + per-level offset ----
  for (int j = tid; j < NPROP; j += bs) {
    float x1 = X1[j], y1 = Y1[j], x2 = X2[j], y2 = Y2[j];
    float minx = fminf(fmaxf(x1, 0.f), Wimg);
    float miny = fminf(fmaxf(y1, 0.f), Himg);
    float maxx = fminf(fmaxf(x2, 0.f), Wimg);
    float maxy = fminf(fmaxf(y2, 0.f), Himg);
    bool keep = (maxx - minx >= MIN_SIZE) && (maxy - miny >= MIN_SIZE);
    float off = (float)(j / NUM_PRE) * pmax;   // level id = slot/2000
    X1[j] = x1 + off; Y1[j] = y1 + off; X2[j] = x2 + off; Y2[j] = y2 + off;
    if (!keep) S[j] = NEGV;
  }
  __syncthreads();

  // ---- 1000 sequential greedy-NMS steps ----
  float* ob = out + (size_t)b * NUM_POST * 4;
  for (int t = 0; t < NUM_POST; ++t) {
    // block argmax over S (tie -> lowest index, like jnp.argmax)
    float bv = NEGV; int bi = 0x7fffffff;
    for (int j = tid; j < NPROP; j += bs) {
      float s = S[j];
      if (s > bv) { bv = s; bi = j; }
    }
    wave_argmax(bv, bi);
    if (lane == 0) { redV[wid] = bv; redI[wid] = bi; }
    __syncthreads();
    if (wid == 0) {
      float v = (lane < nw) ? redV[lane] : NEGV;
      int   i = (lane < nw) ? redI[lane] : 0x7fffffff;
      wave_argmax(v, i);
      if (lane == 0) { ctrl[0] = v; ctrl[1] = __int_as_float(i); }
    }
    __syncthreads();
    const float bestV = ctrl[0];
    const int   bestI = __float_as_int(ctrl[1]);

    if (!(bestV > NEGV)) {               // exhausted: zero remaining outputs
      for (int idx = t * 4 + tid; idx < NUM_POST * 4; idx += bs) ob[idx] = 0.f;
      break;                             // uniform across block
    }

    const float bx1 = X1[bestI], by1 = Y1[bestI];
    const float bx2 = X2[bestI], by2 = Y2[bestI];
    if (tid == 0) {
      ob[t * 4 + 0] = bx1; ob[t * 4 + 1] = by1;
      ob[t * 4 + 2] = bx2; ob[t * 4 + 3] = by2;
      S[bestI] = NEGV;                   // winner leaves the pool
    }
    const float ai = (bx2 - bx1) * (by2 - by1);

    // suppression sweep (each thread owns a strided chunk of S)
    for (int j = tid; j < NPROP; j += bs) {
      float s = S[j];
      if (s > NEGV) {
        float ix1 = fmaxf(bx1, X1[j]);
        float iy1 = fmaxf(by1, Y1[j]);
        float ix2 = fminf(bx2, X2[j]);
        float iy2 = fminf(by2, Y2[j]);
        float inter = fmaxf(ix2 - ix1, 0.f) * fmaxf(iy2 - iy1, 0.f);
        float aj = (X2[j] - X1[j]) * (Y2[j] - Y1[j]);
        float iou = inter / (ai + aj - inter);
        if (iou > NMS_THR) S[j] = NEGV;
      }
    }
    __syncthreads();
  }
}

// ============================================================================
// host-side launcher
// d_in order (setup_inputs insertion order):
//   [cls0,regs0,anch0, cls1,regs1,anch1, cls2,regs2,anch2, cls3,regs3,anch3,
//    image_h, image_w]
// ============================================================================
extern "C" void kernel_launch(void* const* d_in, const int* in_sizes, int n_in,
                              void* d_out, int out_size, void* d_ws, size_t ws_size,
                              hipStream_t stream) {
  (void)in_sizes; (void)n_in; (void)out_size; (void)ws_size;

  LevelArgs la;
  const int Hs[NLEV] = {256, 128, 64, 32};
  for (int l = 0; l < NLEV; ++l) {
    la.cls[l]  = (const float*)d_in[3 * l + 0];
    la.regs[l] = (const float*)d_in[3 * l + 1];
    la.anch[l] = (const float*)d_in[3 * l + 2];
    int sh = 0; while ((1 << sh) < Hs[l] * Hs[l]) ++sh;
    la.hwshift[l] = sh;
  }
  const int* imh = (const int*)d_in[12];
  const int* imw = (const int*)d_in[13];

  float* sc_ws  = (float*)d_ws;               // [16][8000]
  float* box_ws = sc_ws + NBATCH * NPROP;     // [16][8000][4]  (2.56 MB total)

  // one block per (image, level): 64 blocks, all levels in flight at once
  rpn_topk_decode<<<NBATCH * NLEV, 1024, 0, stream>>>(la, sc_ws, box_ws);

  const size_t smem = (size_t)(5 * NPROP + 96) * sizeof(float);  // ~160 KB
  (void)hipFuncSetAttribute(reinterpret_cast<const void*>(rpn_nms),
                            hipFuncAttributeMaxDynamicSharedMemorySize,
                            (int)smem);
  rpn_nms<<<NBATCH, 1024, smem, stream>>>(sc_ws, box_ws, imh, imw,
                                          (float*)d_out);
}